// GCN_10668698763799
// MI455X (gfx1250) — compile-verified
//
#include <hip/hip_runtime.h>

#define HID     256
#define NGRAPH  64
#define BN_EPS  1e-5f

typedef __attribute__((ext_vector_type(2))) float v2f;
typedef __attribute__((ext_vector_type(8))) float v8f;

// ---------------------------------------------------------------- utilities
__global__ void k_zero(float* __restrict__ p, int n) {
  int t = blockIdx.x * blockDim.x + threadIdx.x;
  if (t < n) p[t] = 0.0f;
}

__global__ void k_deg_init(float* __restrict__ deg, int n) {
  int t = blockIdx.x * blockDim.x + threadIdx.x;
  if (t < n) deg[t] = 1.0f;                 // self loop
}

__global__ void k_deg_edges(const int* __restrict__ dst, float* __restrict__ deg, int ne) {
  int e = blockIdx.x * blockDim.x + threadIdx.x;
  if (e < ne) unsafeAtomicAdd(&deg[dst[e]], 1.0f);
}

__global__ void k_dinv(float* __restrict__ deg, int n) {
  int t = blockIdx.x * blockDim.x + threadIdx.x;
  if (t < n) deg[t] = rsqrtf(deg[t]);       // deg >= 1 always
}

// ------------------------------------------------- layer 1: rank-1 "GEMM"
__global__ void k_l1(const float* __restrict__ x, const float* __restrict__ W1,
                     float* __restrict__ xw, int nh) {
  int t = blockIdx.x * blockDim.x + threadIdx.x;
  if (t < nh) xw[t] = x[t >> 8] * W1[t & (HID - 1)];
}

// ------------------------------------------------- fp32 WMMA GEMM  C = A*B
// A: [nrows, 256] row-major, B: [256, 256] row-major, C: [nrows, 256].
// Block = 256 threads = 8 waves, one 16-row M-tile per block.
// A tile (16 KB) staged into LDS via async Global->LDS DMA (ASYNCcnt path).
// Wave w computes N-tiles {2w, 2w+1}, sharing its A fragment.
__global__ void k_gemm(const float* __restrict__ A, const float* __restrict__ B,
                       float* __restrict__ C, int nrows) {
  __shared__ __align__(16) float sA[16 * HID];   // 16 KB
  const int row0 = blockIdx.x * 16;

  if (row0 + 16 <= nrows) {
    // Async DMA: each thread owns one b128 (16 B); INST_OFFSET advances both
    // the LDS and global addresses, so 4 offsets cover the 16 KB tile.
    unsigned lds    = (unsigned)(uintptr_t)sA + threadIdx.x * 16u;
    const float* g  = A + (size_t)row0 * HID + threadIdx.x * 4;
    asm volatile(
        "global_load_async_to_lds_b128 %0, %1, off\n\t"
        "global_load_async_to_lds_b128 %0, %1, off offset:4096\n\t"
        "global_load_async_to_lds_b128 %0, %1, off offset:8192\n\t"
        "global_load_async_to_lds_b128 %0, %1, off offset:12288\n\t"
        "s_wait_asynccnt 0x0"
        :: "v"(lds), "v"(g) : "memory");
  } else {
    for (int i = threadIdx.x; i < 16 * HID; i += 256) {
      int gr = row0 + (i >> 8);
      sA[i] = (gr < nrows) ? A[(size_t)gr * HID + (i & (HID - 1))] : 0.0f;
    }
  }
  __syncthreads();

  const int wave  = threadIdx.x >> 5;
  const int lane  = threadIdx.x & 31;
  const int lh    = lane >> 4;              // lane half: 0 or 1
  const int lm    = lane & 15;
  const int n0    = (wave * 2) * 16;        // first N-tile column
  const int n1    = n0 + 16;

  v8f c0 = {};
  v8f c1 = {};
  for (int kk = 0; kk < HID; kk += 4) {
    const int k0 = kk + 2 * lh;             // this lane's K pair
    v2f a;                                  // A(m=lm, k0..k0+1)
    a.x = sA[lm * HID + k0];
    a.y = sA[lm * HID + k0 + 1];
    v2f b0, b1;                             // B(k0..k0+1, n)
    b0.x = B[(size_t)k0 * HID + n0 + lm];
    b0.y = B[(size_t)(k0 + 1) * HID + n0 + lm];
    b1.x = B[(size_t)k0 * HID + n1 + lm];
    b1.y = B[(size_t)(k0 + 1) * HID + n1 + lm];
    c0 = __builtin_amdgcn_wmma_f32_16x16x4_f32(false, a, false, b0,
                                               (short)0, c0, false, false);
    c1 = __builtin_amdgcn_wmma_f32_16x16x4_f32(false, a, false, b1,
                                               (short)0, c1, false, false);
  }

  // C layout: vgpr r -> M=r (lanes 0-15) / M=r+8 (lanes 16-31)
  float* Cp = C + (size_t)(row0 + 8 * lh) * HID + lm;
  if (row0 + 16 <= nrows) {
#pragma unroll
    for (int r = 0; r < 8; ++r) {
      Cp[r * HID + n0] = c0[r];
      Cp[r * HID + n1] = c1[r];
    }
  } else {
    for (int r = 0; r < 8; ++r) {
      if (row0 + r + 8 * lh < nrows) {
        Cp[r * HID + n0] = c0[r];
        Cp[r * HID + n1] = c1[r];
      }
    }
  }
}

// ------------------------------------------------- aggregation
// agg[i,:] = dinv[i]^2 * xw[i,:] + bias   (self-loop message + bias)
__global__ void k_agg_init(const float* __restrict__ xw, const float* __restrict__ dinv,
                           const float* __restrict__ bias, float* __restrict__ agg, int nh) {
  int t = blockIdx.x * blockDim.x + threadIdx.x;
  if (t >= nh) return;
  float di = dinv[t >> 8];
  agg[t] = di * di * xw[t] + bias[t & (HID - 1)];
}

// agg[dst,:] += dinv[src]*dinv[dst] * xw[src,:]   (64 threads/edge, 4 cols each)
__global__ void k_agg_edges(const int* __restrict__ src, const int* __restrict__ dst,
                            const float* __restrict__ dinv, const float* __restrict__ xw,
                            float* __restrict__ agg, int ne) {
  int t = blockIdx.x * blockDim.x + threadIdx.x;
  int e = t >> 6;
  if (e >= ne) return;
  int cg = (t & 63) << 2;                   // column group of 4
  int s = src[e], d = dst[e];
  float coef = dinv[s] * dinv[d];
  const float4 v = *reinterpret_cast<const float4*>(xw + (size_t)s * HID + cg);
  float* o = agg + (size_t)d * HID + cg;
  unsafeAtomicAdd(o + 0, coef * v.x);
  unsafeAtomicAdd(o + 1, coef * v.y);
  unsafeAtomicAdd(o + 2, coef * v.z);
  unsafeAtomicAdd(o + 3, coef * v.w);
}

// ------------------------------------------------- batch norm (+ReLU)
__global__ void k_bn_stats(const float* __restrict__ x, float* __restrict__ sums,
                           float* __restrict__ sumsq, int nrows) {
  int col = threadIdx.x;                    // 256 threads = 256 columns
  int r0 = blockIdx.x * 256;
  int r1 = r0 + 256 < nrows ? r0 + 256 : nrows;
  float s = 0.0f, sq = 0.0f;
  for (int r = r0; r < r1; ++r) {
    float v = x[(size_t)r * HID + col];
    s += v;
    sq += v * v;
  }
  unsafeAtomicAdd(&sums[col], s);
  unsafeAtomicAdd(&sumsq[col], sq);
}

__global__ void k_bn_apply(float* __restrict__ xio, const float* __restrict__ sums,
                           const float* __restrict__ sumsq, const float* __restrict__ gamma,
                           const float* __restrict__ beta, int nrows, int nh) {
  int t = blockIdx.x * blockDim.x + threadIdx.x;
  if (t >= nh) return;
  int j = t & (HID - 1);
  float inv_n = 1.0f / (float)nrows;
  float mu  = sums[j] * inv_n;
  float var = sumsq[j] * inv_n - mu * mu;
  float y = (xio[t] - mu) * rsqrtf(var + BN_EPS) * gamma[j] + beta[j];
  xio[t] = fmaxf(y, 0.0f);                  // fused ReLU
}

// ------------------------------------------------- pooling + final linear
__global__ void k_pool(const float* __restrict__ h, const int* __restrict__ batch,
                       float* __restrict__ psum, int nh) {
  int t = blockIdx.x * blockDim.x + threadIdx.x;
  if (t >= nh) return;
  int g = batch[t >> 8];
  unsafeAtomicAdd(&psum[g * HID + (t & (HID - 1))], h[t]);
}

__global__ void k_cnt(const int* __restrict__ batch, float* __restrict__ pcnt, int n) {
  int t = blockIdx.x * blockDim.x + threadIdx.x;
  if (t < n) unsafeAtomicAdd(&pcnt[batch[t]], 1.0f);
}

__global__ void k_final(const float* __restrict__ psum, const float* __restrict__ pcnt,
                        const float* __restrict__ lw, const float* __restrict__ lb,
                        float* __restrict__ out) {
  int t = threadIdx.x;                      // 128 = 64 graphs x 2 outputs
  int g = t >> 1, j = t & 1;
  float invc = 1.0f / fmaxf(pcnt[g], 1.0f);
  float acc = 0.0f;
  for (int k = 0; k < HID; ++k) acc += psum[g * HID + k] * lw[k * 2 + j];
  out[g * 2 + j] = acc * invc + lb[j];
}

// ---------------------------------------------------------------- launcher
extern "C" void kernel_launch(void* const* d_in, const int* in_sizes, int n_in,
                              void* d_out, int out_size, void* d_ws, size_t ws_size,
                              hipStream_t stream) {
  const float* x     = (const float*)d_in[0];
  const int*   ei    = (const int*)d_in[1];
  const int*   batch = (const int*)d_in[2];
  const float* W1  = (const float*)d_in[3];
  const float* b1  = (const float*)d_in[4];
  const float* g1  = (const float*)d_in[5];
  const float* be1 = (const float*)d_in[6];
  const float* W2  = (const float*)d_in[7];
  const float* b2  = (const float*)d_in[8];
  const float* g2  = (const float*)d_in[9];
  const float* be2 = (const float*)d_in[10];
  const float* W3  = (const float*)d_in[11];
  const float* b3  = (const float*)d_in[12];
  const float* g3  = (const float*)d_in[13];
  const float* be3 = (const float*)d_in[14];
  const float* lw  = (const float*)d_in[15];
  const float* lb  = (const float*)d_in[16];

  const int N = in_sizes[0];                // 50000
  const int E = in_sizes[1] / 2;            // 800000
  const int* src = ei;
  const int* dst = ei + E;

  float* ws    = (float*)d_ws;
  size_t NH    = (size_t)N * HID;
  float* bufP  = ws;                        // features (layer input / BN output)
  float* bufQ  = ws + NH;                   // xW
  float* dinv  = ws + 2 * NH;               // N (deg -> dinv in place)
  float* sums  = dinv + N;                  // 256
  float* sumsq = sums + HID;                // 256
  float* psum  = sumsq + HID;               // 64*256
  float* pcnt  = psum + NGRAPH * HID;       // 64

  const int nh    = (int)NH;
  const dim3 B(256);
  const int gN    = (N + 255) / 256;
  const int gNH   = (nh + 255) / 256;
  const int gE    = (E + 255) / 256;
  const int gE64  = (int)(((size_t)E * 64 + 255) / 256);
  const int gGemm = (N + 15) / 16;
  const int gBN   = (N + 255) / 256;

  // degrees -> dinv
  k_deg_init<<<gN, B, 0, stream>>>(dinv, N);
  k_deg_edges<<<gE, B, 0, stream>>>(dst, dinv, E);
  k_dinv<<<gN, B, 0, stream>>>(dinv, N);

  // ---- layer 1 (x is [N,1] -> rank-1 product)
  k_l1<<<gNH, B, 0, stream>>>(x, W1, bufQ, nh);
  k_agg_init<<<gNH, B, 0, stream>>>(bufQ, dinv, b1, bufP, nh);
  k_agg_edges<<<gE64, B, 0, stream>>>(src, dst, dinv, bufQ, bufP, E);
  k_zero<<<2, B, 0, stream>>>(sums, 2 * HID);
  k_bn_stats<<<gBN, B, 0, stream>>>(bufP, sums, sumsq, N);
  k_bn_apply<<<gNH, B, 0, stream>>>(bufP, sums, sumsq, g1, be1, N, nh);

  // ---- layers 2 and 3 (WMMA GEMM)
  const float* Ws[2]  = {W2, W3};
  const float* bs[2]  = {b2, b3};
  const float* gs[2]  = {g2, g3};
  const float* bes[2] = {be2, be3};
  for (int l = 0; l < 2; ++l) {
    k_gemm<<<gGemm, B, 0, stream>>>(bufP, Ws[l], bufQ, N);
    k_agg_init<<<gNH, B, 0, stream>>>(bufQ, dinv, bs[l], bufP, nh);
    k_agg_edges<<<gE64, B, 0, stream>>>(src, dst, dinv, bufQ, bufP, E);
    k_zero<<<2, B, 0, stream>>>(sums, 2 * HID);
    k_bn_stats<<<gBN, B, 0, stream>>>(bufP, sums, sumsq, N);
    k_bn_apply<<<gNH, B, 0, stream>>>(bufP, sums, sumsq, gs[l], bes[l], N, nh);
  }

  // ---- mean pool + final linear
  k_zero<<<(NGRAPH * HID + NGRAPH + 255) / 256, B, 0, stream>>>(psum, NGRAPH * HID + NGRAPH);
  k_pool<<<gNH, B, 0, stream>>>(bufP, batch, psum, nh);
  k_cnt<<<gN, B, 0, stream>>>(batch, pcnt, N);
  k_final<<<1, 128, 0, stream>>>(psum, pcnt, lw, lb, (float*)d_out);
}